// QuantumGenerative_45251775431187
// MI455X (gfx1250) — compile-verified
//
#include <hip/hip_runtime.h>
#include <hip/hip_bf16.h>

typedef __attribute__((ext_vector_type(2))) float v2f;
typedef __attribute__((ext_vector_type(8))) float v8f;

#define NQ      12
#define DIM     4096          // 2^12
#define HALFDIM 2048
#define NTHREADS 128
#define NGATES  132           // 10 layers * 12 + final 12

// pad LDS index: stride-64 rows land in distinct banks
__device__ __forceinline__ int PAD(int i) { return i + (i >> 6); }

__global__ __launch_bounds__(NTHREADS)
void quantum_sel_kernel(const float* __restrict__ inputs,
                        const float* __restrict__ w0,
                        const float* __restrict__ w_last,
                        float* __restrict__ out)
{
    __shared__ float sre[4160];          // padded 4096
    __shared__ float sim[4160];
    __shared__ float gm[NGATES * 8];     // per-gate 2x2 complex: 00r,00i,01r,01i,10r,10i,11r,11i

    const int tid = threadIdx.x;
    const int b   = blockIdx.x;

    // ---- phase 0: zero state + precompute all Rot gate matrices ----
    for (int i = tid; i < DIM; i += NTHREADS) {
        int a = PAD(i);
        sre[a] = 0.0f;
        sim[a] = 0.0f;
    }
    for (int g = tid; g < NGATES; g += NTHREADS) {
        float phi, theta, omega;
        if (g < 120) {
            const float* p = w0 + g * 3;      // w0 is (10,12,3) contiguous, g = l*12+i
            phi = p[0]; theta = p[1]; omega = p[2];
        } else {
            const float* p = w_last + (g - 120) * 3;
            phi = p[0]; theta = p[1]; omega = p[2];
        }
        float c = cosf(0.5f * theta), s = sinf(0.5f * theta);
        float ap = 0.5f * (phi + omega), am = 0.5f * (phi - omega);
        float ear =  cosf(ap), eai = -sinf(ap);   // ea = exp(-i(phi+omega)/2)
        float ebr =  cosf(am), ebi =  sinf(am);   // eb = exp(+i(phi-omega)/2)
        float* u = gm + g * 8;
        u[0] =  ear * c;  u[1] =  eai * c;        // u00 = ea*c
        u[2] = -ebr * s;  u[3] = -ebi * s;        // u01 = -eb*s
        u[4] =  ebr * s;  u[5] = -ebi * s;        // u10 = conj(eb)*s
        u[6] =  ear * c;  u[7] = -eai * c;        // u11 = conj(ea)*c
    }
    __syncthreads();

    // ---- embedding: RY(x0) on wire0, RY(x1) on wire1 applied to |0..0> ----
    if (tid < 4) {
        float x0 = inputs[2 * b + 0], x1 = inputs[2 * b + 1];
        float c0 = cosf(0.5f * x0), s0 = sinf(0.5f * x0);
        float c1 = cosf(0.5f * x1), s1 = sinf(0.5f * x1);
        int b0 = tid >> 1, b1 = tid & 1;
        int idx = b0 * 2048 + b1 * 1024;          // wire0 -> bit 11, wire1 -> bit 10
        sre[PAD(idx)] = (b0 ? s0 : c0) * (b1 ? s1 : c1);
    }
    __syncthreads();

    // ---- gate sequence ----
    for (int l = 0; l < 11; ++l) {
        // 12 Rot gates
        for (int w = 0; w < NQ; ++w) {
            int g = (l < 10) ? (l * 12 + w) : (120 + w);
            const float* u = gm + g * 8;
            float u00r = u[0], u00i = u[1], u01r = u[2], u01i = u[3];
            float u10r = u[4], u10i = u[5], u11r = u[6], u11i = u[7];
            int p = 11 - w;                       // bit position of wire w
            int stride = 1 << p;
            for (int t = tid; t < HALFDIM; t += NTHREADS) {
                int i0 = ((t >> p) << (p + 1)) | (t & (stride - 1));
                int a0 = PAD(i0), a1 = PAD(i0 + stride);
                float ar = sre[a0], ai = sim[a0];
                float br = sre[a1], bi = sim[a1];
                sre[a0] = u00r * ar - u00i * ai + u01r * br - u01i * bi;
                sim[a0] = u00r * ai + u00i * ar + u01r * bi + u01i * br;
                sre[a1] = u10r * ar - u10i * ai + u11r * br - u11i * bi;
                sim[a1] = u10r * ai + u10i * ar + u11r * bi + u11i * br;
            }
            __syncthreads();
        }
        // fused CZ ring (only for the 10 entangling layers)
        if (l < 10) {
            int r = l % 11 + 1;
            for (int idx = tid; idx < DIM; idx += NTHREADS) {
                int cnt = 0;
                #pragma unroll
                for (int i = 0; i < NQ; ++i) {
                    int j = i + r; if (j >= NQ) j -= NQ;
                    cnt += (idx >> (11 - i)) & (idx >> (11 - j)) & 1;
                }
                if (cnt & 1) {
                    int a = PAD(idx);
                    sre[a] = -sre[a];
                    sim[a] = -sim[a];
                }
            }
            __syncthreads();
        }
    }

    // ---- RDM via WMMA f32 16x16x4: rho = P P^H, P = psi[a, o(16), r(64)] ----
    // K = a*64 + r (256 total). Wave wv handles a = wv (K in [wv*64, wv*64+64)).
    // A layout (16x4 f32): VGPR v, lane(half,m) -> P[m][base + v + 2*half]
    // B layout (4x16 f32): identical per-lane content for the Gram product.
    const int wv   = tid >> 5;
    const int lane = tid & 31;
    const int half = lane >> 4;
    const int m    = lane & 15;
    const int base = wv * 1040 + m * 65;          // PAD(wv*1024 + m*64)

    v8f accRe = {};
    v8f accIm = {};
    for (int kk = 0; kk < 16; ++kk) {
        int r0 = kk * 4 + 2 * half;
        v2f aRe = { sre[base + r0], sre[base + r0 + 1] };
        v2f aIm = { sim[base + r0], sim[base + r0 + 1] };
        v2f aReN = -aRe;
        // rho_re += Pre*Pre^T + Pim*Pim^T
        accRe = __builtin_amdgcn_wmma_f32_16x16x4_f32(false, aRe, false, aRe,
                                                      (short)0, accRe, false, false);
        accRe = __builtin_amdgcn_wmma_f32_16x16x4_f32(false, aIm, false, aIm,
                                                      (short)0, accRe, false, false);
        // rho_im += Pim*Pre^T - Pre*Pim^T
        accIm = __builtin_amdgcn_wmma_f32_16x16x4_f32(false, aIm, false, aRe,
                                                      (short)0, accIm, false, false);
        accIm = __builtin_amdgcn_wmma_f32_16x16x4_f32(false, aReN, false, aIm,
                                                      (short)0, accIm, false, false);
    }
    __syncthreads();                  // all waves done reading the state

    // stash per-wave partials in (now dead) state LDS: [wv][comp][vgpr][lane]
    #pragma unroll
    for (int v = 0; v < 8; ++v) {
        sre[wv * 512 +       v * 32 + lane] = accRe[v];
        sre[wv * 512 + 256 + v * 32 + lane] = accIm[v];
    }
    __syncthreads();

    // cross-wave reduce + write complex64-interleaved output (b,16,16)
    float* ob = out + (size_t)b * 512;
    for (int s = tid; s < 512; s += NTHREADS) {
        float sum = sre[s] + sre[s + 512] + sre[s + 1024] + sre[s + 1536];
        int comp = s >> 8;            // 0 = re, 1 = im
        int v    = (s >> 5) & 7;
        int ln   = s & 31;
        int M    = v + 8 * (ln >> 4); // C/D layout: row
        int N    = ln & 15;           // col
        ob[(M * 16 + N) * 2 + comp] = sum;
    }
}

extern "C" void kernel_launch(void* const* d_in, const int* in_sizes, int n_in,
                              void* d_out, int out_size, void* d_ws, size_t ws_size,
                              hipStream_t stream) {
    const float* inputs = (const float*)d_in[0];   // (2048, 2) f32
    const float* w0     = (const float*)d_in[1];   // (10, 12, 3) f32
    const float* w_last = (const float*)d_in[2];   // (12, 3) f32
    float* out = (float*)d_out;                    // (2048,16,16) complex64 -> interleaved f32
    (void)in_sizes; (void)n_in; (void)out_size; (void)d_ws; (void)ws_size;
    quantum_sel_kernel<<<2048, NTHREADS, 0, stream>>>(inputs, w0, w_last, out);
}